// GATProteinsModel_36867999269113
// MI455X (gfx1250) — compile-verified
//
#include <hip/hip_runtime.h>
#include <math.h>

#define NN 50000          // nodes
#define NE 500000         // raw edges
#define ET (NE + NN)      // edges + self loops = 550000
#define NF 128            // input features
#define DD 144            // hidden = 18*8 = final out dim
#define NH 8
#define HC 18
#define NG 64             // graphs
#define BNEPS 1e-5f
#define SLOPE 0.2f

typedef __attribute__((ext_vector_type(2))) float v2f;
typedef __attribute__((ext_vector_type(8))) float v8f;

// ---------------- WMMA f32 GEMM: Y[N,M] = X[N,K] @ W[K,M] (+bias) ----------------
// Block = 8 waves = 128-row slab. Loop over column tiles; each iteration stages the
// K x 16 weight panel in LDS (shared by all 8 waves), then every wave runs the
// 16x16x4 f32 WMMA chain for its row tile. A: contiguous b64 loads from global
// (L2-resident), B: ds_load from LDS. EXEC all-1s around WMMA (wave-uniform guard).
__global__ void k_wmma_gemm_f32(const float* __restrict__ X, const float* __restrict__ W,
                                const float* __restrict__ bias, float* __restrict__ Y,
                                int Nrow, int K, int Mcol) {
  __shared__ float sB[DD * 16];                 // up to 144x16 panel = 9.2 KB
  const int lane = threadIdx.x & 31;
  const int wv   = threadIdx.x >> 5;            // 0..7
  const int rowTile = blockIdx.x * 8 + wv;
  const bool active = (rowTile * 16 < Nrow);
  const int r  = lane & 15;
  const int kb = (lane >> 4) << 1;              // 0 (lanes 0-15) or 2 (lanes 16-31)
  const int mb = (lane >> 4) << 3;              // output rows 0..7 or 8..15
  const int tilesM = Mcol >> 4;

  for (int ct = 0; ct < tilesM; ++ct) {
    // cooperative stage of W[:, ct*16 .. ct*16+15]
    for (int idx = threadIdx.x; idx < K * 16; idx += 256) {
      int kk = idx >> 4, cc = idx & 15;
      sB[idx] = W[(size_t)kk * Mcol + ct * 16 + cc];
    }
    __syncthreads();
    if (active) {
      const float* xr = X + (size_t)(rowTile * 16 + r) * K;
      v8f acc = {};
      for (int k = 0; k < K; k += 4) {
        v2f a, b;
        a[0] = xr[k + kb];
        a[1] = xr[k + kb + 1];
        b[0] = sB[(k + kb) * 16 + r];
        b[1] = sB[(k + kb + 1) * 16 + r];
        acc = __builtin_amdgcn_wmma_f32_16x16x4_f32(false, a, false, b, (short)0, acc,
                                                    false, false);
      }
      const int col = ct * 16 + r;
      const float bv = bias ? bias[col] : 0.0f;
      for (int i = 0; i < 8; ++i)
        Y[(size_t)(rowTile * 16 + mb + i) * Mcol + col] = acc[i] + bv;
    }
    __syncthreads();
  }
}

// ---------------- helpers ----------------
__device__ __forceinline__ unsigned encf(float f) {
  unsigned u = __float_as_uint(f);
  return (u & 0x80000000u) ? ~u : (u | 0x80000000u);   // order-preserving
}
__device__ __forceinline__ float decf(unsigned u) {
  u = (u & 0x80000000u) ? (u & 0x7fffffffu) : ~u;
  return __uint_as_float(u);
}
__device__ __forceinline__ void edge_sd(const int* __restrict__ ei, int e, int& s, int& d) {
  if (e < NE) { s = ei[e]; d = ei[NE + e]; }
  else        { s = e - NE; d = s; }                    // self loop
}
__device__ __forceinline__ float lrelu(float v) { return v > 0.f ? v : SLOPE * v; }

// ---------------- attention logits ----------------
__global__ void k_attn_logits(const float* __restrict__ hp, const float* __restrict__ a_s,
                              const float* __restrict__ a_d, float* __restrict__ als,
                              float* __restrict__ ald, int H, int C) {
  int t = blockIdx.x * blockDim.x + threadIdx.x;
  if (t >= NN * H) return;
  int n = t / H, h = t - n * H;
  const float* v = hp + (size_t)n * (H * C) + h * C;
  float s0 = 0.f, s1 = 0.f;
  for (int c = 0; c < C; ++c) { s0 += v[c] * a_s[h * C + c]; s1 += v[c] * a_d[h * C + c]; }
  als[t] = s0; ald[t] = s1;
}

// ---------------- segment max over dst ----------------
__global__ void k_edge_max(const int* __restrict__ ei, const float* __restrict__ als,
                           const float* __restrict__ ald, unsigned* __restrict__ mmax, int H) {
  int t = blockIdx.x * blockDim.x + threadIdx.x;
  if (t >= ET * H) return;
  int e = t / H, h = t - e * H;
  int s, d; edge_sd(ei, e, s, d);
  float v = lrelu(als[s * H + h] + ald[d * H + h]);
  atomicMax(&mmax[d * H + h], encf(v));
}

// ---------------- exp + segment denominator ----------------
__global__ void k_edge_exp(const int* __restrict__ ei, const float* __restrict__ als,
                           const float* __restrict__ ald, const unsigned* __restrict__ mmax,
                           float* __restrict__ ex, float* __restrict__ den, int H) {
  int t = blockIdx.x * blockDim.x + threadIdx.x;
  if (t >= ET * H) return;
  int e = t / H, h = t - e * H;
  int s, d; edge_sd(ei, e, s, d);
  float v = lrelu(als[s * H + h] + ald[d * H + h]);
  float ev = expf(v - decf(mmax[d * H + h]));
  ex[t] = ev;
  atomicAdd(&den[d * H + h], ev);
}

// ---------------- normalize: ex -> alpha (per edge, head) ----------------
__global__ void k_alpha(const int* __restrict__ ei, float* __restrict__ ex,
                        const float* __restrict__ den, int H) {
  int t = blockIdx.x * blockDim.x + threadIdx.x;
  if (t >= ET * H) return;
  int e = t / H, h = t - e * H;
  int s, d; edge_sd(ei, e, s, d);
  ex[t] = ex[t] / (den[d * H + h] + 1e-16f);
}

// ---------------- weighted scatter (float4): out[dst,:] += alpha * hp[src,:] -----
__global__ void k_edge_scatter(const int* __restrict__ ei, const float* __restrict__ hp,
                               const float* __restrict__ alpha, float* __restrict__ out,
                               int H, int C) {
  const int Dq = DD / 4;                                // 36 quads per edge
  int t = blockIdx.x * blockDim.x + threadIdx.x;
  if (t >= ET * Dq) return;
  int e = t / Dq, q = t - e * Dq;
  int dc = q * 4;
  int s, d; edge_sd(ei, e, s, d);
  const float4 hv = *(const float4*)(hp + (size_t)s * DD + dc);
  const float* al = alpha + (size_t)e * H;
  float a0 = al[(dc    ) / C];
  float a1 = al[(dc + 1) / C];
  float a2 = al[(dc + 2) / C];
  float a3 = al[(dc + 3) / C];
  float* o = out + (size_t)d * DD + dc;
  atomicAdd(o + 0, hv.x * a0);
  atomicAdd(o + 1, hv.y * a1);
  atomicAdd(o + 2, hv.z * a2);
  atomicAdd(o + 3, hv.w * a3);
}

// ---------------- bias + relu (float4) ----------------
__global__ void k_bias_relu(float* __restrict__ X, const float* __restrict__ b,
                            int totalq, int Dd) {
  int t = blockIdx.x * blockDim.x + threadIdx.x;
  if (t >= totalq) return;
  int j = (t * 4) % Dd;
  float4 v = *(float4*)(X + t * 4);
  v.x = fmaxf(v.x + b[j + 0], 0.f);
  v.y = fmaxf(v.y + b[j + 1], 0.f);
  v.z = fmaxf(v.z + b[j + 2], 0.f);
  v.w = fmaxf(v.w + b[j + 3], 0.f);
  *(float4*)(X + t * 4) = v;
}

// ---------------- batchnorm stats (biased var), one block per column -------------
__global__ void k_bn_stats(const float* __restrict__ X, int N, int Dd,
                           float* __restrict__ mu, float* __restrict__ rstd) {
  int j = blockIdx.x;
  float s = 0.f, s2 = 0.f;
  for (int n = threadIdx.x; n < N; n += blockDim.x) {
    float v = X[(size_t)n * Dd + j]; s += v; s2 += v * v;
  }
  __shared__ float sh[256], sh2[256];
  sh[threadIdx.x] = s; sh2[threadIdx.x] = s2;
  __syncthreads();
  for (int st = blockDim.x >> 1; st > 0; st >>= 1) {
    if ((int)threadIdx.x < st) { sh[threadIdx.x] += sh[threadIdx.x + st];
                                 sh2[threadIdx.x] += sh2[threadIdx.x + st]; }
    __syncthreads();
  }
  if (threadIdx.x == 0) {
    float m = sh[0] / N;
    float var = sh2[0] / N - m * m;
    mu[j] = m; rstd[j] = rsqrtf(var + BNEPS);
  }
}

// ---------------- BN apply (float4), optional relu ----------------
__global__ void k_bn_apply(const float* __restrict__ X, float* __restrict__ Y,
                           const float* __restrict__ mu, const float* __restrict__ rstd,
                           const float* __restrict__ g, const float* __restrict__ b,
                           int totalq, int Dd, int doRelu) {
  int t = blockIdx.x * blockDim.x + threadIdx.x;
  if (t >= totalq) return;
  int j = (t * 4) % Dd;
  float4 v = *(const float4*)(X + t * 4);
  float4 o;
  o.x = g[j + 0] * (v.x - mu[j + 0]) * rstd[j + 0] + b[j + 0];
  o.y = g[j + 1] * (v.y - mu[j + 1]) * rstd[j + 1] + b[j + 1];
  o.z = g[j + 2] * (v.z - mu[j + 2]) * rstd[j + 2] + b[j + 2];
  o.w = g[j + 3] * (v.w - mu[j + 3]) * rstd[j + 3] + b[j + 3];
  if (doRelu) {
    o.x = fmaxf(o.x, 0.f); o.y = fmaxf(o.y, 0.f);
    o.z = fmaxf(o.z, 0.f); o.w = fmaxf(o.w, 0.f);
  }
  *(float4*)(Y + t * 4) = o;
}

// ---------------- mean pool per graph (float4 gather, 4 atomics) ----------------
__global__ void k_pool_sum(const float* __restrict__ X, const int* __restrict__ batch,
                           float* __restrict__ gsum) {
  const int Dq = DD / 4;
  int t = blockIdx.x * blockDim.x + threadIdx.x;
  if (t >= NN * Dq) return;
  int n = t / Dq, q = t - n * Dq;
  float4 v = *(const float4*)(X + (size_t)n * DD + q * 4);
  float* o = gsum + batch[n] * DD + q * 4;
  atomicAdd(o + 0, v.x); atomicAdd(o + 1, v.y);
  atomicAdd(o + 2, v.z); atomicAdd(o + 3, v.w);
}
__global__ void k_pool_cnt(const int* __restrict__ batch, float* __restrict__ cnt) {
  int n = blockIdx.x * blockDim.x + threadIdx.x;
  if (n >= NN) return;
  atomicAdd(&cnt[batch[n]], 1.0f);
}
__global__ void k_pool_div(float* __restrict__ gsum, const float* __restrict__ cnt) {
  int t = blockIdx.x * blockDim.x + threadIdx.x;
  if (t >= NG * DD) return;
  gsum[t] /= fmaxf(cnt[t / DD], 1.0f);
}

// ---------------- small dense linear (MLP head) ----------------
__global__ void k_linear(const float* __restrict__ X, const float* __restrict__ W,
                         const float* __restrict__ b, float* __restrict__ Y,
                         int N, int K, int M) {
  int t = blockIdx.x * blockDim.x + threadIdx.x;
  if (t >= N * M) return;
  int n = t / M, m = t - n * M;
  float s = b[m];
  for (int k = 0; k < K; ++k) s += X[n * K + k] * W[k * M + m];
  Y[t] = s;
}

// =================================================================================
static inline int cdiv(long long a, int b) { return (int)((a + b - 1) / b); }

extern "C" void kernel_launch(void* const* d_in, const int* in_sizes, int n_in,
                              void* d_out, int out_size, void* d_ws, size_t ws_size,
                              hipStream_t stream) {
  const float* x   = (const float*)d_in[0];
  const int*   ei  = (const int*)  d_in[1];
  const int*   bat = (const int*)  d_in[2];
  const float* We  = (const float*)d_in[3];
  const float* be  = (const float*)d_in[4];
  const float* Wg  = (const float*)d_in[5];   // [3,144,144]
  const float* asg = (const float*)d_in[6];   // [3,8,18]
  const float* adg = (const float*)d_in[7];
  const float* bg  = (const float*)d_in[8];   // [3,144]
  const float* gam = (const float*)d_in[9];
  const float* bet = (const float*)d_in[10];
  const float* W3  = (const float*)d_in[11];
  const float* as3 = (const float*)d_in[12];
  const float* ad3 = (const float*)d_in[13];
  const float* b3  = (const float*)d_in[14];
  const float* gm3 = (const float*)d_in[15];
  const float* bt3 = (const float*)d_in[16];
  const float* M1  = (const float*)d_in[17];
  const float* bm1 = (const float*)d_in[18];
  const float* gm1 = (const float*)d_in[19];
  const float* bb1 = (const float*)d_in[20];
  const float* M2  = (const float*)d_in[21];
  const float* bm2 = (const float*)d_in[22];
  const float* gm2 = (const float*)d_in[23];
  const float* bb2 = (const float*)d_in[24];
  const float* M3  = (const float*)d_in[25];
  const float* bm3 = (const float*)d_in[26];

  // workspace carve-up (floats); total ~27.6M floats ~= 111 MB
  float* w = (float*)d_ws;
  const size_t NBUF = (size_t)NN * DD;           // 7.2M
  float*    bufA = w;                 w += NBUF;  // layer input / BN output
  float*    hp   = w;                 w += NBUF;  // per-head transformed feats
  float*    agg  = w;                 w += NBUF;  // aggregated output
  float*    als  = w;                 w += (size_t)NN * NH;
  float*    ald  = w;                 w += (size_t)NN * NH;
  unsigned* mmax = (unsigned*)w;      w += (size_t)NN * NH;
  float*    den  = w;                 w += (size_t)NN * NH;
  float*    ex   = w;                 w += (size_t)ET * NH;  // exp -> alpha in place
  float*    mu   = w;                 w += 256;
  float*    rstd = w;                 w += 256;
  float*    gsum = w;                 w += NG * DD;
  float*    cnt  = w;                 w += NG;
  float*    h1   = w;                 w += NG * 72;
  float*    h2   = w;                 w += NG * 36;

  const int B = 256;
  const int gemmBlocks = cdiv(NN / 16, 8);       // 8 row-tiles per block

  // ---- encoder: bufA = x @ We + be  (50000x128 @ 128x144)
  k_wmma_gemm_f32<<<gemmBlocks, B, 0, stream>>>(x, We, be, bufA, NN, NF, DD);

  // ---- 3 multi-head GAT layers + final single-head layer
  for (int L = 0; L < 4; ++L) {
    const int H = (L < 3) ? NH : 1;
    const int C = (L < 3) ? HC : DD;
    const float* Wl = (L < 3) ? (Wg + (size_t)L * DD * DD) : W3;
    const float* as = (L < 3) ? (asg + (size_t)L * NH * HC) : as3;
    const float* ad = (L < 3) ? (adg + (size_t)L * NH * HC) : ad3;
    const float* bl = (L < 3) ? (bg + (size_t)L * DD) : b3;
    const float* gl = (L < 3) ? (gam + (size_t)L * DD) : gm3;
    const float* tl = (L < 3) ? (bet + (size_t)L * DD) : bt3;

    k_wmma_gemm_f32<<<gemmBlocks, B, 0, stream>>>(bufA, Wl, nullptr, hp, NN, DD, DD);
    k_attn_logits<<<cdiv((long long)NN * H, B), B, 0, stream>>>(hp, as, ad, als, ald, H, C);

    hipMemsetAsync(mmax, 0, (size_t)NN * H * 4, stream);   // 0 < encf(any float)
    hipMemsetAsync(den,  0, (size_t)NN * H * 4, stream);
    k_edge_max<<<cdiv((long long)ET * H, B), B, 0, stream>>>(ei, als, ald, mmax, H);
    k_edge_exp<<<cdiv((long long)ET * H, B), B, 0, stream>>>(ei, als, ald, mmax, ex, den, H);
    k_alpha<<<cdiv((long long)ET * H, B), B, 0, stream>>>(ei, ex, den, H);

    hipMemsetAsync(agg, 0, NBUF * 4, stream);
    k_edge_scatter<<<cdiv((long long)ET * (DD / 4), B), B, 0, stream>>>(ei, hp, ex, agg, H, C);

    k_bias_relu<<<cdiv((long long)(NBUF / 4), B), B, 0, stream>>>(agg, bl, (int)(NBUF / 4), DD);
    k_bn_stats<<<DD, B, 0, stream>>>(agg, NN, DD, mu, rstd);
    k_bn_apply<<<cdiv((long long)(NBUF / 4), B), B, 0, stream>>>(agg, bufA, mu, rstd, gl, tl,
                                                                 (int)(NBUF / 4), DD, 0);
  }

  // ---- global mean pool
  hipMemsetAsync(gsum, 0, NG * DD * 4, stream);
  hipMemsetAsync(cnt,  0, NG * 4, stream);
  k_pool_sum<<<cdiv((long long)NN * (DD / 4), B), B, 0, stream>>>(bufA, bat, gsum);
  k_pool_cnt<<<cdiv(NN, B), B, 0, stream>>>(bat, cnt);
  k_pool_div<<<cdiv(NG * DD, B), B, 0, stream>>>(gsum, cnt);

  // ---- MLP head: lin -> BN -> relu, twice, then plain lin
  k_linear<<<cdiv(NG * 72, B), B, 0, stream>>>(gsum, M1, bm1, h1, NG, DD, 72);
  k_bn_stats<<<72, B, 0, stream>>>(h1, NG, 72, mu, rstd);
  k_bn_apply<<<cdiv(NG * 72 / 4, B), B, 0, stream>>>(h1, h1, mu, rstd, gm1, bb1,
                                                     NG * 72 / 4, 72, 1);

  k_linear<<<cdiv(NG * 36, B), B, 0, stream>>>(h1, M2, bm2, h2, NG, 72, 36);
  k_bn_stats<<<36, B, 0, stream>>>(h2, NG, 36, mu, rstd);
  k_bn_apply<<<cdiv(NG * 36 / 4, B), B, 0, stream>>>(h2, h2, mu, rstd, gm2, bb2,
                                                     NG * 36 / 4, 36, 1);

  k_linear<<<cdiv(NG * 112, B), B, 0, stream>>>(h2, M3, bm3, (float*)d_out, NG, 36, 112);
}